// GNNConv_34153579938136
// MI455X (gfx1250) — compile-verified
//
#include <hip/hip_runtime.h>
#include <hip/hip_bf16.h>

typedef __attribute__((ext_vector_type(2))) float v2f;
typedef __attribute__((ext_vector_type(8))) float v8f;

#define DIN   256
#define DOUT  64
#define KCHUNK 128
#define LDS_STRIDE (KCHUNK + 4)   // 132: row stride mod 64 banks = 4 -> conflict-free v2f reads

__device__ __forceinline__ void atomicAddF(float* p, float v) {
    __hip_atomic_fetch_add(p, v, __ATOMIC_RELAXED, __HIP_MEMORY_SCOPE_AGENT);
}

// ---------------- degree / norm ----------------
__global__ void fill_deg_kernel(float* deg, int n) {
    int i = blockIdx.x * blockDim.x + threadIdx.x;
    if (i < n) deg[i] = 1.0f;                       // self loop
}

__global__ void deg_count_kernel(const long long* __restrict__ dst, float* deg, int e) {
    int i = blockIdx.x * blockDim.x + threadIdx.x;
    if (i < e) atomicAddF(&deg[(int)dst[i]], 1.0f);
}

__global__ void dinv_kernel(float* deg, int n) {
    int i = blockIdx.x * blockDim.x + threadIdx.x;
    if (i < n) deg[i] = rsqrtf(deg[i]);             // deg >= 1 always
}

// ---------------- fp32 WMMA GEMM: h = xs @ W^T ----------------
__global__ __launch_bounds__(256) void gcn_gemm_wmma_kernel(
    const float* __restrict__ xs, const float* __restrict__ W,
    float* __restrict__ h, int n)
{
    __shared__ float sW[DOUT * LDS_STRIDE];         // 33,792 B

    const int tid  = threadIdx.x;
    const int wave = tid >> 5;
    const int lane = tid & 31;
    const int m    = lane & 15;                     // row within A-tile / col within B-tile
    const int kh   = lane >> 4;                     // K-half select (0/1)
    const int rowBase = blockIdx.x * 128 + wave * 16;

    int row = rowBase + m;
    if (row >= n) row = n - 1;                      // clamp loads; stores guarded below
    const float* xrow = xs + (long)row * DIN;

    v8f acc0 = {}, acc1 = {}, acc2 = {}, acc3 = {};

    for (int kk = 0; kk < DIN; kk += KCHUNK) {
        __syncthreads();                            // protect previous-chunk readers
        // stage W[:, kk:kk+128] into LDS (8192 floats, 2048 float4 / 256 threads)
        for (int i = tid; i < DOUT * (KCHUNK / 4); i += 256) {
            const int r  = i / (KCHUNK / 4);
            const int c4 = i % (KCHUNK / 4);
            const float4 w4 = *(const float4*)(&W[(long)r * DIN + kk + c4 * 4]);
            float* dp = &sW[r * LDS_STRIDE + c4 * 4];
            dp[0] = w4.x; dp[1] = w4.y; dp[2] = w4.z; dp[3] = w4.w;
        }
        __syncthreads();

        #pragma unroll 4
        for (int k = 0; k < KCHUNK; k += 4) {
            const int ka = k + kh * 2;              // ISA A/B layout: K = 2*(lane>>4) + {0,1}
            v2f a = *(const v2f*)(&xrow[kk + ka]);
            v2f b0 = *(const v2f*)(&sW[( 0 + m) * LDS_STRIDE + ka]);
            v2f b1 = *(const v2f*)(&sW[(16 + m) * LDS_STRIDE + ka]);
            v2f b2 = *(const v2f*)(&sW[(32 + m) * LDS_STRIDE + ka]);
            v2f b3 = *(const v2f*)(&sW[(48 + m) * LDS_STRIDE + ka]);
            acc0 = __builtin_amdgcn_wmma_f32_16x16x4_f32(false, a, false, b0, (short)0, acc0, false, false);
            acc1 = __builtin_amdgcn_wmma_f32_16x16x4_f32(false, a, false, b1, (short)0, acc1, false, false);
            acc2 = __builtin_amdgcn_wmma_f32_16x16x4_f32(false, a, false, b2, (short)0, acc2, false, false);
            acc3 = __builtin_amdgcn_wmma_f32_16x16x4_f32(false, a, false, b3, (short)0, acc3, false, false);
        }
    }

    // D layout: VGPR r -> M = r + 8*(lane>>4), N = lane&15
    #pragma unroll
    for (int r = 0; r < 8; ++r) {
        const int orow = rowBase + r + kh * 8;
        if (orow < n) {
            float* op = h + (long)orow * DOUT + m;
            op[ 0] = acc0[r];
            op[16] = acc1[r];
            op[32] = acc2[r];
            op[48] = acc3[r];
        }
    }
}

// ---------------- self-loop init: acc = h * dinv^2 ----------------
__global__ void init_acc_kernel(const float* __restrict__ h, const float* __restrict__ dinv,
                                float* __restrict__ acc, int n)
{
    int i = blockIdx.x * blockDim.x + threadIdx.x;
    if (i < n * DOUT) {
        const int node = i >> 6;
        const float di = dinv[node];
        acc[i] = h[i] * di * di;
    }
}

// ---------------- edge scatter: 64 threads per edge ----------------
__global__ void edge_scatter_kernel(const long long* __restrict__ ei,
                                    const float* __restrict__ h,
                                    const float* __restrict__ dinv,
                                    float* __restrict__ acc, int e)
{
    const long g = (long)blockIdx.x * blockDim.x + threadIdx.x;
    const long edge = g >> 6;
    const int  c    = (int)(g & 63);
    if (edge < e) {
        const int s = (int)ei[edge];
        const int d = (int)ei[(long)e + edge];
        const float nrm = dinv[s] * dinv[d];
        atomicAddF(&acc[(long)d * DOUT + c], h[(long)s * DOUT + c] * nrm);
    }
}

// ---------------- LayerNorm: one wave32 per node, 2 feats/lane ----------------
__global__ __launch_bounds__(256) void layernorm_kernel(
    const float* __restrict__ acc, const float* __restrict__ bias,
    const float* __restrict__ gamma, const float* __restrict__ beta,
    float* __restrict__ out, int n)
{
    const int wave = threadIdx.x >> 5;
    const int lane = threadIdx.x & 31;
    const int node = blockIdx.x * 8 + wave;
    if (node >= n) return;

    float2 v  = *(const float2*)(&acc[(long)node * DOUT + lane * 2]);
    float2 bb = *(const float2*)(&bias[lane * 2]);
    v.x += bb.x; v.y += bb.y;

    float s = v.x + v.y;
    #pragma unroll
    for (int off = 16; off; off >>= 1) s += __shfl_xor(s, off, 32);
    const float mu = s * (1.0f / 64.0f);

    const float dx = v.x - mu, dy = v.y - mu;
    float q = dx * dx + dy * dy;
    #pragma unroll
    for (int off = 16; off; off >>= 1) q += __shfl_xor(q, off, 32);
    const float rstd = rsqrtf(q * (1.0f / 64.0f) + 1e-5f);

    float2 g  = *(const float2*)(&gamma[lane * 2]);
    float2 be = *(const float2*)(&beta[lane * 2]);
    float2 o;
    o.x = dx * rstd * g.x + be.x;
    o.y = dy * rstd * g.y + be.y;
    *(float2*)(&out[(long)node * DOUT + lane * 2]) = o;
}

extern "C" void kernel_launch(void* const* d_in, const int* in_sizes, int n_in,
                              void* d_out, int out_size, void* d_ws, size_t ws_size,
                              hipStream_t stream) {
    const float*     xs    = (const float*)d_in[0];
    const long long* ei    = (const long long*)d_in[1];   // int64 [2, E]
    const float*     W     = (const float*)d_in[2];
    const float*     bias  = (const float*)d_in[3];
    const float*     gamma = (const float*)d_in[4];
    const float*     beta  = (const float*)d_in[5];
    float*           out   = (float*)d_out;

    const int N = in_sizes[0] / DIN;
    const int E = in_sizes[1] / 2;

    char* ws = (char*)d_ws;
    float* dinv = (float*)ws;
    size_t off1 = (((size_t)N * 4) + 255) & ~(size_t)255;
    float* h    = (float*)(ws + off1);
    size_t off2 = off1 + ((((size_t)N * DOUT * 4) + 255) & ~(size_t)255);
    float* acc  = (float*)(ws + off2);

    fill_deg_kernel   <<<(N + 255) / 256, 256, 0, stream>>>(dinv, N);
    deg_count_kernel  <<<(E + 255) / 256, 256, 0, stream>>>(ei + E, dinv, E);
    dinv_kernel       <<<(N + 255) / 256, 256, 0, stream>>>(dinv, N);

    gcn_gemm_wmma_kernel<<<(N + 127) / 128, 256, 0, stream>>>(xs, W, h, N);

    init_acc_kernel   <<<((long)N * DOUT + 255) / 256, 256, 0, stream>>>(h, dinv, acc, N);

    const long scat_threads = (long)E * 64;
    edge_scatter_kernel<<<(unsigned)((scat_threads + 255) / 256), 256, 0, stream>>>(ei, h, dinv, acc, E);

    layernorm_kernel  <<<(N + 7) / 8, 256, 0, stream>>>(acc, bias, gamma, beta, out, N);
}